// RiskInfo_70325794504826
// MI455X (gfx1250) — compile-verified
//
#include <hip/hip_runtime.h>
#include <math.h>

typedef __attribute__((ext_vector_type(2))) float v2f;
typedef __attribute__((ext_vector_type(8))) float v8f;

#define D_CA    12
#define D_COURT 4
#define D_CATE  4
#define D_TOT   20
#define CHUNK   32            // events per thread in the accumulation phase
#define M_TILES_PER_WAVE 8    // 16-row M tiles handled by each wave in projection

// ---------------------------------------------------------------------------
// Phase 0: zero the per-company accumulator in workspace
// ---------------------------------------------------------------------------
__global__ void risk_zero_acc(float* __restrict__ acc, int n) {
    int i = blockIdx.x * blockDim.x + threadIdx.x;
    int stride = gridDim.x * blockDim.x;
    for (; i < n; i += stride) acc[i] = 0.0f;
}

// ---------------------------------------------------------------------------
// Phase 1: gather embeddings, apply decay, run-compressed segment scatter-add.
// seg_ids are sorted, so each thread keeps a register accumulator for the
// current company and only flushes (20 f32 atomics) when the id changes.
// ---------------------------------------------------------------------------
__global__ void risk_accumulate(const int*   __restrict__ cause_idx,
                                const int*   __restrict__ court_idx,
                                const int*   __restrict__ cate_idx,
                                const int*   __restrict__ seg_ids,
                                const float* __restrict__ time_interval,
                                const float* __restrict__ ca_table,     // [200,12]
                                const float* __restrict__ court_table,  // [50,4]
                                const float* __restrict__ cate_table,   // [10,4]
                                const float* __restrict__ decay_w,      // scalar
                                float*       __restrict__ acc,          // [n_comp,20]
                                int n_events) {
    const float dw = decay_w[0];
    const float4* __restrict__ ca4 = (const float4*)ca_table;     // rows 16B aligned (48B stride)
    const float4* __restrict__ co4 = (const float4*)court_table;  // 16B stride
    const float4* __restrict__ ct4 = (const float4*)cate_table;   // 16B stride

    const int tid  = blockIdx.x * blockDim.x + threadIdx.x;
    const int base = tid * CHUNK;
    if (base >= n_events) return;

    float a[D_TOT];
    int cur = -1;

    auto flush = [&]() {
#pragma unroll
        for (int j = 0; j < D_TOT; ++j)
            unsafeAtomicAdd(&acc[cur * D_TOT + j], a[j]);
    };

    auto process = [&](int s, int ic, int io, int it, float t) {
        const float dec = dw / logf(2.7182818284590452354f + t);

        const float4 r0 = ca4[ic * 3 + 0];
        const float4 r1 = ca4[ic * 3 + 1];
        const float4 r2 = ca4[ic * 3 + 2];
        const float4 r3 = co4[io];
        const float4 r4 = ct4[it];

        float x[D_TOT];
        x[0]=r0.x;  x[1]=r0.y;  x[2]=r0.z;  x[3]=r0.w;
        x[4]=r1.x;  x[5]=r1.y;  x[6]=r1.z;  x[7]=r1.w;
        x[8]=r2.x;  x[9]=r2.y;  x[10]=r2.z; x[11]=r2.w;
        x[12]=r3.x; x[13]=r3.y; x[14]=r3.z; x[15]=r3.w;
        x[16]=r4.x; x[17]=r4.y; x[18]=r4.z; x[19]=r4.w;

        if (s != cur) {
            if (cur >= 0) flush();
            cur = s;
#pragma unroll
            for (int j = 0; j < D_TOT; ++j) a[j] = dec * x[j];
        } else {
#pragma unroll
            for (int j = 0; j < D_TOT; ++j) a[j] = fmaf(dec, x[j], a[j]);
        }
    };

    if (base + CHUNK <= n_events) {
        // fast path: 4 events per vector load packet (16B-aligned: base % 4 == 0)
        for (int i = 0; i < CHUNK; i += 4) {
            const int e = base + i;
            const int4   sg = *(const int4*)  (seg_ids       + e);
            const int4   ca = *(const int4*)  (cause_idx     + e);
            const int4   co = *(const int4*)  (court_idx     + e);
            const int4   ct = *(const int4*)  (cate_idx      + e);
            const float4 tt = *(const float4*)(time_interval + e);
            process(sg.x, ca.x, co.x, ct.x, tt.x);
            process(sg.y, ca.y, co.y, ct.y, tt.y);
            process(sg.z, ca.z, co.z, ct.z, tt.z);
            process(sg.w, ca.w, co.w, ct.w, tt.w);
        }
    } else {
        for (int i = 0; i < CHUNK; ++i) {
            const int e = base + i;
            if (e >= n_events) break;
            process(seg_ids[e], cause_idx[e], court_idx[e], cate_idx[e],
                    time_interval[e]);
        }
    }
    if (cur >= 0) flush();
}

// ---------------------------------------------------------------------------
// Phase 2: out[m, o] = sum_k acc[m, k] * W[o, k]   via V_WMMA_F32_16X16X4_F32
//   K = 20 = 5 exact chunks of 4; N = 20 -> two 16-wide N tiles (2nd padded).
//   One wave per block; B fragments hoisted and reused across 8 M tiles.
//   Full tiles: C/D staged in LDS, drained as 5 coalesced unguarded b64
//   stores/lane. Single-wave WG => LDS is in-order within the wave, so no
//   barriers are needed: __builtin_amdgcn_wave_barrier() is a zero-cost
//   scheduling fence and the backend inserts the precise s_wait_dscnt.
// ---------------------------------------------------------------------------
__global__ void risk_project(const float* __restrict__ acc,   // [n_comp,20]
                             const float* __restrict__ w,     // proj_w [20,20] (o-major)
                             float*       __restrict__ out,   // [n_comp,20]
                             int n_comp) {
    __shared__ float stage[16 * 32];    // one wave per block: 2 KB

    const int lane  = threadIdx.x;      // 0..31
    const int half  = lane >> 4;        // 0: lanes 0-15, 1: lanes 16-31
    const int l     = lane & 15;
    const int tile0 = blockIdx.x * M_TILES_PER_WAVE;

    // ---- preload B fragments, shared across all M tiles handled by this wave
    // B 4x16 f32 layout: VGPR j, lane-half h -> K = h*2 + j, N = lane&15
    // B[k][n] = proj_w[n][k]  (y = x @ W^T)
    const int  n1     = 16 + l;
    const bool nvalid = (n1 < D_TOT);
    const int  n1c    = nvalid ? n1 : (D_TOT - 1);   // clamped row, zeroed below

    v2f b0[5], b1[5];
#pragma unroll
    for (int kc = 0; kc < 5; ++kc) {
        const int k = kc * 4 + half * 2;
        const float2 t0 = *(const float2*)(w + l   * D_TOT + k);
        const float2 t1 = *(const float2*)(w + n1c * D_TOT + k);
        b0[kc].x = t0.x;
        b0[kc].y = t0.y;
        b1[kc].x = nvalid ? t1.x : 0.0f;   // v_cndmask, not a masked load
        b1[kc].y = nvalid ? t1.y : 0.0f;
    }

    // ---- lane-constant linear->(row,col) map for the coalesced drain:
    // tile output = 320 contiguous floats; lane writes float2 at p*64 + lane*2
    int saddr[5];
#pragma unroll
    for (int p = 0; p < 5; ++p) {
        const int idx = p * 64 + lane * 2;      // even, 0..318
        const int srow = idx / 20;
        const int scol = idx - srow * 20;       // even -> float2 stays in-row
        saddr[p] = srow * 32 + scol;
    }

    const int tiles_total = (n_comp + 15) >> 4;
    const int full_tiles  = n_comp >> 4;
    int t_end_full = tile0 + M_TILES_PER_WAVE;
    if (t_end_full > full_tiles) t_end_full = full_tiles;
    int t_end = tile0 + M_TILES_PER_WAVE;
    if (t_end > tiles_total) t_end = tiles_total;

    // strength-reduced pointers (advance 16 rows = 320 floats per tile)
    const float* __restrict__ arow = acc + (tile0 * 16 + l) * D_TOT + half * 2;
    float*       __restrict__ dst  = out + tile0 * (16 * D_TOT);

    int tile = tile0;
    for (; tile < t_end_full; ++tile) {
        // A 16x4 f32 layout: lanes 0-15 hold K = kc*4+{0,1}; lanes 16-31 +{2,3}
        v8f c0 = {};   // output columns 0..15
        v8f c1 = {};   // output columns 16..19 (padded to 16)
#pragma unroll
        for (int kc = 0; kc < 5; ++kc) {
            const float2 ta = *(const float2*)(arow + kc * 4);
            v2f a;
            a.x = ta.x;
            a.y = ta.y;
            c0 = __builtin_amdgcn_wmma_f32_16x16x4_f32(false, a, false, b0[kc],
                                                       (short)0, c0, false, false);
            c1 = __builtin_amdgcn_wmma_f32_16x16x4_f32(false, a, false, b1[kc],
                                                       (short)0, c1, false, false);
        }

        // ---- stage C/D in LDS, drain coalesced & unguarded
        // C/D 16x16 f32 layout: VGPR r -> M = r + half*8, N = lane&15
#pragma unroll
        for (int r = 0; r < 8; ++r) {
            stage[(r + half * 8) * 32 + l]      = c0[r];
            stage[(r + half * 8) * 32 + 16 + l] = c1[r];
        }
        __builtin_amdgcn_wave_barrier();   // scheduling fence only; LDS is
                                           // in-order within a wave
#pragma unroll
        for (int p = 0; p < 5; ++p) {
            const float2 v = *(const float2*)&stage[saddr[p]];
            *(float2*)(dst + p * 64 + lane * 2) = v;
        }
        __builtin_amdgcn_wave_barrier();   // keep next tile's LDS stores after
                                           // this tile's LDS loads

        arow += 16 * D_TOT;
        dst  += 16 * D_TOT;
    }

    if (tile < t_end) {
        // ---- partial tail tile: clamped loads + guarded stores (rare)
        const int m  = tile * 16 + l;
        const int mc = (m < n_comp) ? m : (n_comp - 1);
        v8f c0 = {};
        v8f c1 = {};
#pragma unroll
        for (int kc = 0; kc < 5; ++kc) {
            const int k = kc * 4 + half * 2;
            const float2 ta = *(const float2*)(acc + mc * D_TOT + k);
            v2f a;
            a.x = ta.x;
            a.y = ta.y;
            c0 = __builtin_amdgcn_wmma_f32_16x16x4_f32(false, a, false, b0[kc],
                                                       (short)0, c0, false, false);
            c1 = __builtin_amdgcn_wmma_f32_16x16x4_f32(false, a, false, b1[kc],
                                                       (short)0, c1, false, false);
        }
#pragma unroll
        for (int r = 0; r < 8; ++r) {
            const int row = tile * 16 + r + half * 8;
            if (row < n_comp) {
                out[row * D_TOT + l] = c0[r];
                if (nvalid) out[row * D_TOT + n1] = c1[r];
            }
        }
    }
}

// ---------------------------------------------------------------------------
// Host launcher
// ---------------------------------------------------------------------------
extern "C" void kernel_launch(void* const* d_in, const int* in_sizes, int n_in,
                              void* d_out, int out_size, void* d_ws, size_t ws_size,
                              hipStream_t stream) {
    const int*   cause_idx   = (const int*)  d_in[0];
    const int*   court_idx   = (const int*)  d_in[1];
    const int*   cate_idx    = (const int*)  d_in[2];
    const int*   seg_ids     = (const int*)  d_in[3];
    const float* time_interv = (const float*)d_in[4];
    const float* ca_table    = (const float*)d_in[5];
    const float* court_table = (const float*)d_in[6];
    const float* cate_table  = (const float*)d_in[7];
    const float* proj_w      = (const float*)d_in[8];
    const float* decay_w     = (const float*)d_in[9];

    const int n_events = in_sizes[0];
    const int n_comp   = out_size / D_TOT;

    float* acc = (float*)d_ws;             // [n_comp, 20] f32 accumulator
    float* out = (float*)d_out;

    // Phase 0: zero accumulator
    {
        const int n = n_comp * D_TOT;
        const int block = 256;
        int grid = (n + block * 4 - 1) / (block * 4);
        if (grid < 1) grid = 1;
        if (grid > 2048) grid = 2048;
        risk_zero_acc<<<grid, block, 0, stream>>>(acc, n);
    }

    // Phase 1: gather + decay + run-compressed segment sum (pre-projection)
    {
        const int block = 256;
        const int nthreads = (n_events + CHUNK - 1) / CHUNK;
        const int grid = (nthreads + block - 1) / block;
        risk_accumulate<<<grid, block, 0, stream>>>(
            cause_idx, court_idx, cate_idx, seg_ids, time_interv,
            ca_table, court_table, cate_table, decay_w, acc, n_events);
    }

    // Phase 2: WMMA projection acc @ W^T -> out
    {
        const int waves = (n_comp + 16 * M_TILES_PER_WAVE - 1) / (16 * M_TILES_PER_WAVE);
        risk_project<<<waves, 32, 0, stream>>>(acc, proj_w, out, n_comp);
    }
}